// DPSN_6717328851338
// MI455X (gfx1250) — compile-verified
//
#include <hip/hip_runtime.h>
#include <math.h>

#define BATCH 256
#define DIM   768
#define HCN   128
#define HSN   256
#define POOL  100000
#define TOPK  1325
#define NSORT 2048
#define KC    64     // K-chunk staged per TDM transfer

typedef __attribute__((ext_vector_type(2))) float    v2f;
typedef __attribute__((ext_vector_type(8))) float    v8f;
typedef __attribute__((ext_vector_type(4))) unsigned u32x4;
typedef __attribute__((ext_vector_type(8))) int      i32x8;
typedef __attribute__((ext_vector_type(4))) int      i32x4;

// ---- monotone float<->uint key transform (order-preserving) ----
__device__ __forceinline__ unsigned f2k(float f) {
  unsigned u = __float_as_uint(f);
  return (u & 0x80000000u) ? ~u : (u | 0x80000000u);
}
__device__ __forceinline__ float k2f(unsigned k) {
  unsigned u = (k & 0x80000000u) ? (k ^ 0x80000000u) : ~k;
  return __uint_as_float(u);
}

// ============================================================
// K1: complexity head  c = sigmoid(relu(x@Wc1+bc1)@Wc2+bc2)
// ============================================================
__global__ __launch_bounds__(HCN) void complexity_kernel(
    const float* __restrict__ X, const float* __restrict__ Wc1,
    const float* __restrict__ bc1, const float* __restrict__ Wc2,
    const float* __restrict__ bc2, float* __restrict__ comp) {
  __shared__ float xs[DIM];
  __shared__ float red[HCN];
  const int b = blockIdx.x, t = threadIdx.x;
  const float* xr = X + (size_t)b * DIM;
  for (int i = t; i < DIM; i += HCN) xs[i] = xr[i];
  __syncthreads();
  float h = bc1[t];
  for (int kk = 0; kk < DIM; ++kk) h += xs[kk] * Wc1[kk * HCN + t];
  h = fmaxf(h, 0.f);
  red[t] = h * Wc2[t];
  __syncthreads();
  for (int s = HCN >> 1; s > 0; s >>= 1) {
    if (t < s) red[t] += red[t + s];
    __syncthreads();
  }
  if (t == 0) {
    float z = red[0] + bc2[0];
    comp[b] = 1.f / (1.f + expf(-z));
  }
}

// ============================================================
// K2: H = relu(x @ Ws1 + bs1)  [256x256], K=768
// one wave32 per 16x16 tile, V_WMMA_F32_16X16X4_F32
// A frag: lanes 0-15 -> M=lane, K={k0,k0+1}; lanes16-31 -> K={k0+2,k0+3}
// C frag: vgpr v -> M = v + 8*half, N = lane%16
// ============================================================
__global__ __launch_bounds__(32) void gemm_wmma_f32(
    const float* __restrict__ A, const float* __restrict__ Bm,
    const float* __restrict__ bias, float* __restrict__ C,
    int Kdim, int Ndim, int doRelu) {
  const int tn   = blockIdx.x << 4;
  const int tm   = blockIdx.y << 4;
  const int lane = threadIdx.x;
  const int half = lane >> 4;
  const int l16  = lane & 15;
  const float* arow = A + (size_t)(tm + l16) * Kdim;
  v8f acc = {};
  for (int k0 = 0; k0 < Kdim; k0 += 4) {
    const int ka = k0 + (half << 1);
    v2f av, bv;
    av[0] = arow[ka];
    av[1] = arow[ka + 1];
    bv[0] = Bm[(size_t)ka * Ndim + tn + l16];
    bv[1] = Bm[(size_t)(ka + 1) * Ndim + tn + l16];
    acc = __builtin_amdgcn_wmma_f32_16x16x4_f32(
        false, av, false, bv, (short)0, acc, false, false);
  }
  const float bcol = bias[tn + l16];
#pragma unroll
  for (int v = 0; v < 8; ++v) {
    const int m = v + (half << 3);
    float c = acc[v] + bcol;
    if (doRelu) c = fmaxf(c, 0.f);
    C[(size_t)(tm + m) * Ndim + tn + l16] = c;
  }
}

// ============================================================
// K3: scores = H @ Ws2 + bs2   [256 x 100000], K=256
// One block per 16-col N strip; 16 waves = 16 M tiles so Ws2 is
// read from HBM exactly once. B K-chunks (16 x KC f32 = 4 KB) are
// staged into double-buffered LDS by the Tensor Data Mover
// (tensor_load_to_lds, TENSORcnt pipelined), consumed via ds_load
// into V_WMMA_F32_16X16X4_F32.
// ============================================================
__global__ __launch_bounds__(512) void scores_gemm_tdm(
    const float* __restrict__ H, const float* __restrict__ Ws2,
    const float* __restrict__ bs2, float* __restrict__ S) {
  __shared__ float btile[2][KC * 16];   // assumed at LDS offset 0
  const int tn   = blockIdx.x << 4;     // N strip base
  const int t    = threadIdx.x;
  const int wv   = t >> 5;              // wave id = M tile id (0..15)
  const int lane = t & 31;
  const int half = lane >> 4;
  const int l16  = lane & 15;
  const int tm   = wv << 4;

  // ---- issue one TDM 2D tile load: Ws2[kc..kc+KC-1, tn..tn+15] ----
  auto issue = [&](int kc, int buf) {
    const unsigned long long ga =
        (unsigned long long)(size_t)Ws2 +
        ((unsigned long long)kc * POOL + (unsigned)tn) * 4ull;
    u32x4 g0;
    g0[0] = 1u;                                    // count=1, user mode
    g0[1] = (unsigned)buf * (KC * 16 * 4);         // lds_addr (bytes)
    g0[2] = (unsigned)ga;                          // global_addr[31:0]
    g0[3] = (unsigned)((ga >> 32) & 0x01FFFFFFull) // global_addr[56:32]
            | (2u << 30);                          // type=2 (image)
    i32x8 g1;
    g1[0] = (int)(2u << 16);                       // data_size=2 -> 4B
    g1[1] = (int)((POOL & 0xFFFFu) << 16);         // tensor_dim0[15:0]
    g1[2] = (int)((POOL >> 16) | ((HSN & 0xFFFFu) << 16)); // dim0 hi | dim1 lo
    g1[3] = (int)(16u << 16);                      // dim1 hi(0) | tile_dim0=16
    g1[4] = (int)KC;                               // tile_dim1=KC, tile_dim2=0
    g1[5] = (int)POOL;                             // tensor_dim0_stride lo32
    g1[6] = 0;                                     // stride hi | dim1_stride lo
    g1[7] = 0;
    i32x4 gz4 = {0, 0, 0, 0};                      // groups 2/3 unused (2D tile)
    i32x8 gz8 = {0, 0, 0, 0, 0, 0, 0, 0};
    __builtin_amdgcn_tensor_load_to_lds(g0, g1, gz4, gz4, gz8, 0);
  };

  if (wv == 0) issue(0, 0);

  v8f acc = {};
  const float* arow = H + (size_t)(tm + l16) * HSN;

  for (int c = 0; c < HSN / KC; ++c) {
    if (wv == 0) {
      if (c + 1 < HSN / KC) {
        issue((c + 1) * KC, (c + 1) & 1);          // prefetch next chunk
        __builtin_amdgcn_s_wait_tensorcnt(1);      // chunk c landed (in-order)
      } else {
        __builtin_amdgcn_s_wait_tensorcnt(0);
      }
    }
    __syncthreads();
    const float* bt = btile[c & 1];
#pragma unroll
    for (int k0 = 0; k0 < KC; k0 += 4) {
      const int ka = k0 + (half << 1);
      v2f av, bv;
      av[0] = arow[c * KC + ka];                   // A from L2-resident H
      av[1] = arow[c * KC + ka + 1];
      bv[0] = bt[ka * 16 + l16];                   // B from LDS (TDM-staged)
      bv[1] = bt[(ka + 1) * 16 + l16];
      acc = __builtin_amdgcn_wmma_f32_16x16x4_f32(
          false, av, false, bv, (short)0, acc, false, false);
    }
    __syncthreads();
  }

  const float bcol = bs2[tn + l16];
#pragma unroll
  for (int v = 0; v < 8; ++v) {
    const int m = v + (half << 3);
    S[(size_t)(tm + m) * POOL + tn + l16] = acc[v] + bcol;
  }
}

// ============================================================
// K4: exact top-k (4-pass radix select) + bitonic sort + softmax
// ============================================================
__global__ __launch_bounds__(1024) void topk_kernel(
    const float* __restrict__ scores, float* __restrict__ wts,
    int* __restrict__ idxOut, float* __restrict__ idxF) {
  __shared__ unsigned long long skey[NSORT];
  __shared__ unsigned hist[256];
  __shared__ unsigned s_prefix, s_need, s_cnt;
  __shared__ float s_sum;
  const int b = blockIdx.x, t = threadIdx.x;
  const int NT = 1024;
  const float* sc = scores + (size_t)b * POOL;

  if (t == 0) { s_prefix = 0u; s_need = TOPK; }
  __syncthreads();

  for (int pass = 3; pass >= 0; --pass) {
    const int shift = pass << 3;
    const unsigned hmask = (pass == 3) ? 0u : (0xFFFFFFFFu << (shift + 8));
    if (t < 256) hist[t] = 0u;
    __syncthreads();
    const unsigned pfx = s_prefix;
    for (int i = t; i < POOL; i += NT) {
      const unsigned key = f2k(sc[i]);
      if ((key & hmask) == pfx)
        atomicAdd(&hist[(key >> shift) & 255u], 1u);
    }
    __syncthreads();
    if (t == 0) {
      unsigned need = s_need, cum = 0u;
      for (int bk = 255; bk >= 0; --bk) {
        const unsigned c = hist[bk];
        if (cum + c >= need) {
          s_prefix = pfx | ((unsigned)bk << shift);
          s_need   = need - cum;
          break;
        }
        cum += c;
      }
    }
    __syncthreads();
  }
  const unsigned T = s_prefix;

  for (int i = t; i < NSORT; i += NT) skey[i] = 0ull;
  if (t == 0) s_cnt = 0u;
  __syncthreads();
  for (int i = t; i < POOL; i += NT) {
    const unsigned key = f2k(sc[i]);
    if (key >= T) {
      const unsigned pos = atomicAdd(&s_cnt, 1u);
      if (pos < NSORT)
        skey[pos] = ((unsigned long long)key << 32) |
                    (unsigned long long)(0xFFFFFFFFu - (unsigned)i);
    }
  }
  __syncthreads();

  // bitonic sort descending on (key, ~idx): score desc, index asc
  for (unsigned k2 = 2; k2 <= NSORT; k2 <<= 1) {
    for (unsigned j = k2 >> 1; j > 0; j >>= 1) {
      for (unsigned i = (unsigned)t; i < NSORT; i += NT) {
        const unsigned ixj = i ^ j;
        if (ixj > i) {
          const unsigned long long a = skey[i], c = skey[ixj];
          const bool descBlock = ((i & k2) == 0);
          if (descBlock ? (a < c) : (a > c)) { skey[i] = c; skey[ixj] = a; }
        }
      }
      __syncthreads();
    }
  }

  const float m = k2f((unsigned)(skey[0] >> 32));
  if (t == 0) s_sum = 0.f;
  __syncthreads();
  float loc = 0.f;
  for (int i = t; i < TOPK; i += NT)
    loc += expf(k2f((unsigned)(skey[i] >> 32)) - m);
  atomicAdd(&s_sum, loc);
  __syncthreads();
  const float inv = 1.f / s_sum;
  for (int i = t; i < TOPK; i += NT) {
    const unsigned long long e = skey[i];
    const unsigned idx = 0xFFFFFFFFu - (unsigned)(e & 0xFFFFFFFFull);
    const float w = expf(k2f((unsigned)(e >> 32)) - m) * inv;
    wts[(size_t)b * TOPK + i]    = w;
    idxOut[(size_t)b * TOPK + i] = (int)idx;
    idxF[(size_t)b * TOPK + i]   = (float)idx;
  }
}

// ============================================================
// K5: fused gather / dot / tanh / weighted-axpy / residual
// ============================================================
__global__ __launch_bounds__(256) void executor_kernel(
    const float* __restrict__ X, const float* __restrict__ pool,
    const float* __restrict__ wts, const int* __restrict__ idx,
    float* __restrict__ out) {
  __shared__ float xs[DIM];
  __shared__ float accs[DIM];
  const int b = blockIdx.x, t = threadIdx.x;
  const int wv = t >> 5, lane = t & 31;
  const float* xr = X + (size_t)b * DIM;
  for (int i = t; i < DIM; i += 256) { xs[i] = xr[i]; accs[i] = 0.f; }
  __syncthreads();

  float acc[24];
#pragma unroll
  for (int j = 0; j < 24; ++j) acc[j] = 0.f;

  for (int kk = wv; kk < TOPK; kk += 8) {
    const int   r = idx[(size_t)b * TOPK + kk];
    const float w = wts[(size_t)b * TOPK + kk];
    const float* pr = pool + (size_t)r * DIM;
    const int kn = kk + 8;
    if (kn < TOPK) {
      const int rn = idx[(size_t)b * TOPK + kn];
      __builtin_prefetch(pool + (size_t)rn * DIM + lane * 24, 0, 0);
    }
    float pv[24];
    float dot = 0.f;
#pragma unroll
    for (int j = 0; j < 24; ++j) {
      pv[j] = pr[lane + (j << 5)];
      dot += pv[j] * xs[lane + (j << 5)];
    }
#pragma unroll
    for (int off = 16; off > 0; off >>= 1)
      dot += __shfl_xor(dot, off, 32);
    const float a = tanhf(dot) * w;
#pragma unroll
    for (int j = 0; j < 24; ++j) acc[j] += a * pv[j];
  }
#pragma unroll
  for (int j = 0; j < 24; ++j)
    atomicAdd(&accs[lane + (j << 5)], acc[j]);
  __syncthreads();
  for (int i = t; i < DIM; i += 256)
    out[(size_t)b * DIM + i] = xs[i] + accs[i];
}

// ============================================================
extern "C" void kernel_launch(void* const* d_in, const int* in_sizes, int n_in,
                              void* d_out, int out_size, void* d_ws, size_t ws_size,
                              hipStream_t stream) {
  (void)in_sizes; (void)n_in; (void)out_size; (void)ws_size;
  const float* x    = (const float*)d_in[0];
  const float* pool = (const float*)d_in[1];
  const float* Wc1  = (const float*)d_in[2];
  const float* bc1  = (const float*)d_in[3];
  const float* Wc2  = (const float*)d_in[4];
  const float* bc2  = (const float*)d_in[5];
  const float* Ws1  = (const float*)d_in[6];
  const float* bs1  = (const float*)d_in[7];
  const float* Ws2  = (const float*)d_in[8];
  const float* bs2  = (const float*)d_in[9];

  // output: [B,D] residual | [B] complexity | [B,K] indices (as float)
  float* out  = (float*)d_out;
  float* comp = out + (size_t)BATCH * DIM;
  float* idxF = comp + BATCH;

  // workspace: H | scores | weights | int indices
  float* H      = (float*)d_ws;
  float* scores = H + (size_t)BATCH * HSN;
  float* wts    = scores + (size_t)BATCH * POOL;
  int*   idxI   = (int*)(wts + (size_t)BATCH * TOPK);

  complexity_kernel<<<BATCH, HCN, 0, stream>>>(x, Wc1, bc1, Wc2, bc2, comp);

  // H = relu(x @ Ws1 + bs1)   [256 x 256], K=768
  gemm_wmma_f32<<<dim3(HSN / 16, BATCH / 16), 32, 0, stream>>>(
      x, Ws1, bs1, H, DIM, HSN, 1);

  // scores = H @ Ws2 + bs2    [256 x 100000], K=256  (100000 = 6250*16)
  scores_gemm_tdm<<<POOL / 16, 512, 0, stream>>>(H, Ws2, bs2, scores);

  topk_kernel<<<BATCH, 1024, 0, stream>>>(scores, wts, idxI, idxF);

  executor_kernel<<<BATCH, 256, 0, stream>>>(x, pool, wts, idxI, out);
}